// StateSpaceBlock_48954037240297
// MI455X (gfx1250) — compile-verified
//
#include <hip/hip_runtime.h>
#include <math.h>

// Problem dims (match reference)
#define B_ 16
#define S_ 256
#define D_ 512
#define H_ 512
#define N_ 256

typedef __attribute__((ext_vector_type(2))) float v2f;
typedef __attribute__((ext_vector_type(8))) float v8f;

// Compute an NT-wide strip of 16x16 f32 tiles:
//   C[:, j*16:(j+1)*16] = A(16xK, row stride lda) @ B(Kx16*NT, row stride ldb)
// using V_WMMA_F32_16X16X4_F32.  One A fragment feeds NT WMMAs per k-step.
// Per ISA 7.12.2 (32-bit layouts):
//   A frag: lane holds A[lane%16][k + r + 2*(lane/16)]  (r = vgpr 0/1)
//   B frag: lane holds B[k + r + 2*(lane/16)][lane%16]
//   C/D  : lane/vgpr r holds C[r + 8*(lane/16)][lane%16]
template <int NT>
__device__ __forceinline__ void wmma_gemm_strip(const float* __restrict__ A, int lda,
                                                const float* __restrict__ Bm, int ldb,
                                                int K, v8f (&acc)[NT]) {
  const int lane = threadIdx.x & 31;
  const int m  = lane & 15;          // A row / B column within tile
  const int kk = (lane >> 4) << 1;   // 0 for lanes 0-15, 2 for lanes 16-31
#pragma unroll 4
  for (int k = 0; k < K; k += 4) {
    v2f a;
    const float* ap = A + (size_t)m * lda + k + kk;
    a.x = ap[0];                     // K = k+kk   (vgpr 0)
    a.y = ap[1];                     // K = k+kk+1 (vgpr 1)
    const float* bp = Bm + (size_t)(k + kk) * ldb + m;
#pragma unroll
    for (int j = 0; j < NT; ++j) {
      v2f b;
      b.x = bp[j * 16];
      b.y = bp[j * 16 + ldb];
      acc[j] = __builtin_amdgcn_wmma_f32_16x16x4_f32(
          /*neg_a=*/false, a, /*neg_b=*/false, b,
          /*c_mod=*/(short)0, acc[j], /*reuse_a=*/false, /*reuse_b=*/false);
    }
  }
}

// Store one 16x16 D tile (+ per-column bias) with row stride ldc.
__device__ __forceinline__ void store_tile(float* __restrict__ C, int ldc,
                                           const v8f& acc,
                                           const float* __restrict__ bias) {
  const int lane = threadIdx.x & 31;
  const int n  = lane & 15;
  const int mb = (lane >> 4) << 3;   // 0 or 8
  const float bv = bias[n];
#pragma unroll
  for (int r = 0; r < 8; ++r)
    C[(size_t)(mb + r) * ldc + n] = acc[r] + bv;
}

template <int NT>
__device__ __forceinline__ void store_strip(float* __restrict__ C, int ldc,
                                            const v8f (&acc)[NT],
                                            const float* __restrict__ bias) {
#pragma unroll
  for (int j = 0; j < NT; ++j)
    store_tile(C + j * 16, ldc, acc[j], bias + j * 16);
}

// ---------------- Kernel 1: x_proj[(b*S+t)][h] = x @ W_in + b_in ----------------
// rows = B*S = 4096 (256 row tiles), cols = H = 512 -> 8 strips of 64. 2048 waves.
__global__ void k_proj(const float* __restrict__ x, const float* __restrict__ W_in,
                       const float* __restrict__ b_in, float* __restrict__ xp) {
  const int w  = (blockIdx.x * blockDim.x + threadIdx.x) >> 5;
  const int tm = w >> 3;   // row tile (0..255)
  const int ts = w & 7;    // 64-wide strip (0..7)
  v8f acc[4] = {};
  wmma_gemm_strip<4>(x + (size_t)tm * 16 * D_, D_, W_in + ts * 64, H_, D_, acc);
  store_strip<4>(xp + (size_t)tm * 16 * H_ + ts * 64, H_, acc, b_in + ts * 64);
}

// ---------------- Kernel 2: sp[t][b][n] = x_proj[:,t,:] @ W_sp[t] + b_sp[t] ----
// A rows are b=0..15 at xp[b*S*H + t*H] (lda = S*H). N=256 -> 8 strips of 32.
// waves = S * 8 = 2048.
__global__ void k_sp(const float* __restrict__ xp, const float* __restrict__ W_sp,
                     const float* __restrict__ b_sp, float* __restrict__ sp) {
  const int w  = (blockIdx.x * blockDim.x + threadIdx.x) >> 5;
  const int t  = w >> 3;   // timestep (0..255)
  const int ts = w & 7;    // 32-wide strip (0..7)
  v8f acc[2] = {};
  wmma_gemm_strip<2>(xp + (size_t)t * H_, S_ * H_,
                     W_sp + (size_t)t * H_ * N_ + ts * 32, N_, H_, acc);
  store_strip<2>(sp + (size_t)t * B_ * N_ + ts * 32, N_, acc,
                 b_sp + (size_t)t * N_ + ts * 32);
}

// ---------------- Kernel 3: elementwise tanh scan over t ----------------------
// 4096 independent (b,n) chains; fully coalesced per step.
__global__ void k_scan(const float* __restrict__ sp, float* __restrict__ st) {
  const int tid = blockIdx.x * blockDim.x + threadIdx.x;  // 0..4095
  float s = 0.0f;
  for (int t = 0; t < S_; ++t) {
    s = tanhf(s + sp[(size_t)t * (B_ * N_) + tid]);
    st[(size_t)t * (B_ * N_) + tid] = s;
  }
}

// ---------------- Kernel 4: out[b][t][d] = state[t] @ W_out[t] + b_out[t] -----
// D=512 -> 16 strips of 32. waves = S * 16 = 4096. Output row stride = S*D.
__global__ void k_out(const float* __restrict__ st, const float* __restrict__ W_out,
                      const float* __restrict__ b_out, float* __restrict__ out) {
  const int w  = (blockIdx.x * blockDim.x + threadIdx.x) >> 5;
  const int t  = w >> 4;   // timestep (0..255)
  const int ts = w & 15;   // 32-wide strip (0..15)
  v8f acc[2] = {};
  wmma_gemm_strip<2>(st + (size_t)t * B_ * N_, N_,
                     W_out + (size_t)t * N_ * D_ + ts * 32, D_, N_, acc);
  store_strip<2>(out + (size_t)t * D_ + ts * 32, S_ * D_, acc,
                 b_out + (size_t)t * D_ + ts * 32);
}

extern "C" void kernel_launch(void* const* d_in, const int* in_sizes, int n_in,
                              void* d_out, int out_size, void* d_ws, size_t ws_size,
                              hipStream_t stream) {
  const float* x     = (const float*)d_in[0];
  const float* W_in  = (const float*)d_in[1];
  const float* b_in  = (const float*)d_in[2];
  const float* W_sp  = (const float*)d_in[3];
  const float* b_sp  = (const float*)d_in[4];
  const float* W_out = (const float*)d_in[5];
  const float* b_out = (const float*)d_in[6];
  float* out = (float*)d_out;

  // Workspace layout: xp (8 MB) | sp (4 MB) | states (4 MB)
  float* xp = (float*)d_ws;
  float* sp = xp + (size_t)B_ * S_ * H_;
  float* st = sp + (size_t)S_ * B_ * N_;

  // 8 waves (256 threads) per block throughout.
  k_proj<<<2048 / 8, 256, 0, stream>>>(x, W_in, b_in, xp);
  k_sp  <<<2048 / 8, 256, 0, stream>>>(xp, W_sp, b_sp, sp);
  k_scan<<<16,       256, 0, stream>>>(sp, st);
  k_out <<<4096 / 8, 256, 0, stream>>>(st, W_out, b_out, out);
}